// GaussianSTEQuantizer_22471268893156
// MI455X (gfx1250) — compile-verified
//
#include <hip/hip_runtime.h>
#include <stdint.h>

// GaussianSTEQuantizer for MI455X (gfx1250).
// out[r,c] = levels[searchsorted(midpoints, x[r,c]/std_r, 'left')] * std_r
// std_r = sqrt(mean(x[r,:]^2)) + 1e-8, row length 2048.
//
// HBM-bound: 1.07 GB traffic -> ~46us at 23.3 TB/s. Keep VALU under the
// issue roofline:
//   - CDNA5 async global->LDS row staging (ASYNCcnt + s_wait_asynccnt)
//   - wave32 shuffle-tree RMS reduction (deterministic)
//   - per-block 256-cell quantization LUT in LDS (std is uniform per row):
//     per element = 1 fma + clamp + cvt + ds_load_b128 + cmp + cndmask
//   - non-temporal float4 stores (write-once stream; spare the 192MB L2)

#define ROW_LEN   2048
#define NTHREADS  256           // 8 wave32 waves
#define HALF      1024          // floats, second half of row
#define LUT_N     256

typedef float v4f __attribute__((ext_vector_type(4)));   // native vector for b128 + NT store

__global__ __launch_bounds__(NTHREADS)
void gauss_ste_quant_kernel(const float* __restrict__ x,
                            const float* __restrict__ levels,
                            float* __restrict__ out,
                            int nrows)
{
    __shared__ __align__(16) float row[ROW_LEN];
    __shared__ __align__(16) v4f   lut[LUT_N];    // {scaledBnd, sLvlLo, sLvlHi, 0}
    __shared__ float lvls[16];
    __shared__ float nbnd[15];                    // normalized midpoint boundaries
    __shared__ float wsum[NTHREADS / 32];

    const int tid = threadIdx.x;
    const int rid = blockIdx.x;
    if (rid >= nrows) return;

    const float* gsrc = x   + (size_t)rid * ROW_LEN;
    float*       gdst = out + (size_t)rid * ROW_LEN;

    // ---- CDNA5 async copy: global -> LDS (2 x b128 per lane = full 8KiB row) ----
    {
        unsigned l0 = (unsigned)(uintptr_t)(&row[tid * 4]);
        unsigned l1 = (unsigned)(uintptr_t)(&row[HALF + tid * 4]);
        unsigned long long g0 = (unsigned long long)(uintptr_t)(gsrc + tid * 4);
        unsigned long long g1 = (unsigned long long)(uintptr_t)(gsrc + HALF + tid * 4);
        asm volatile("global_load_async_to_lds_b128 %0, %1, off"
                     :: "v"(l0), "v"(g0) : "memory");
        asm volatile("global_load_async_to_lds_b128 %0, %1, off"
                     :: "v"(l1), "v"(g1) : "memory");
    }

    if (tid < 16) lvls[tid] = levels[tid];

    asm volatile("s_wait_asynccnt 0x0" ::: "memory");
    __syncthreads();

    // normalized boundaries (consumed after the next barrier)
    if (tid < 15) nbnd[tid] = 0.5f * (lvls[tid] + lvls[tid + 1]);

    // ---- pass 1: sum of squares (8 elements / thread, from LDS) ----
    v4f a = *(const v4f*)(&row[tid * 4]);
    v4f b = *(const v4f*)(&row[HALF + tid * 4]);
    float ss = a.x*a.x + a.y*a.y + a.z*a.z + a.w*a.w
             + b.x*b.x + b.y*b.y + b.z*b.z + b.w*b.w;

    #pragma unroll
    for (int m = 16; m >= 1; m >>= 1)
        ss += __shfl_xor(ss, m, 32);
    if ((tid & 31) == 0) wsum[tid >> 5] = ss;
    __syncthreads();

    float tot = 0.0f;
    #pragma unroll
    for (int i = 0; i < NTHREADS / 32; ++i) tot += wsum[i];

    const float stdv = sqrtf(tot * (1.0f / (float)ROW_LEN)) + 1e-8f;

    // ---- build per-block quantization LUT (std uniform across the row) ----
    // Grid over normalized space: cell j = [g0 + j*w, g0 + (j+1)*w).
    // Cell width (~0.02) << min boundary gap (~0.26) => <=1 boundary per cell.
    const float blo = nbnd[0], bhi = nbnd[14];
    const float w   = (bhi - blo) * (1.0f / (float)(LUT_N - 2));
    const float g0  = blo - w;
    {
        const float cl = g0 + w * (float)tid;     // cell left edge (normalized)
        const float cr = cl + w;                  // cell right edge
        int k = 0;                                // #boundaries strictly < cl
        #pragma unroll
        for (int i = 0; i < 15; ++i) k += (nbnd[i] < cl) ? 1 : 0;
        const int   kc     = (k < 15) ? k : 14;
        const float bk     = nbnd[kc];
        const bool  inCell = (k < 15) && (bk < cr);
        const float sLo    = lvls[(k < 16) ? k : 15] * stdv;
        const float sHi    = inCell ? lvls[k + 1] * stdv : sLo;
        const float sB     = inCell ? bk * stdv : 3.0e38f;
        lut[tid] = (v4f){sB, sLo, sHi, 0.0f};
    }
    __syncthreads();

    // ---- pass 2: quantize (7 VALU + 1 ds_load_b128 per element) ----
    const float invW = 1.0f / w;
    const float A = (1.0f / stdv) * invW;   // v*A + B == (v/std - g0)/w
    const float B = -g0 * invW;

    auto quant = [&](float v) -> float {
        float jf = fminf(fmaxf(v * A + B, 0.0f), (float)(LUT_N - 1));
        int j = (int)jf;
        v4f e = lut[j];
        return (v > e.x) ? e.z : e.y;       // strict cmp == searchsorted 'left'
    };

    v4f oa, ob;
    oa.x = quant(a.x); oa.y = quant(a.y); oa.z = quant(a.z); oa.w = quant(a.w);
    ob.x = quant(b.x); ob.y = quant(b.y); ob.z = quant(b.z); ob.w = quant(b.w);

    // ---- non-temporal coalesced 16B stores (write-once stream) ----
    __builtin_nontemporal_store(oa, (v4f*)(gdst + tid * 4));
    __builtin_nontemporal_store(ob, (v4f*)(gdst + HALF + tid * 4));
}

extern "C" void kernel_launch(void* const* d_in, const int* in_sizes, int n_in,
                              void* d_out, int out_size, void* d_ws, size_t ws_size,
                              hipStream_t stream) {
    const float* x      = (const float*)d_in[0];   // (8, 8192, 2048) f32
    const float* levels = (const float*)d_in[1];   // (16,) f32 sorted
    float* out = (float*)d_out;                    // same shape as x, f32

    const int nrows = in_sizes[0] / ROW_LEN;       // 65536 rows
    dim3 grid((unsigned)nrows), block(NTHREADS);
    hipLaunchKernelGGL(gauss_ste_quant_kernel, grid, block, 0, stream,
                       x, levels, out, nrows);
}